// Data_Augmentation_Dummy_4647154615610
// MI455X (gfx1250) — compile-verified
//
#include <hip/hip_runtime.h>
#include <hip/hip_bf16.h>

typedef __attribute__((ext_vector_type(2))) float v2f;
typedef __attribute__((ext_vector_type(8))) float v8f;

#define DIM   512
#define HALF  256
#define TILES_PER_IMG 1024   // (512/16) * (512/16)
#define WAVES_PER_BLOCK 8    // 256 threads = 8 wave32

// 12-byte RGB pixel; 4-byte aligned -> global_load/store_b96 path
struct __align__(4) Pix3 { float r, g, b; };

__global__ __launch_bounds__(256)
void Data_Augmentation_warp_kernel(
    const float* __restrict__ images,
    const float* __restrict__ rot_n, const float* __restrict__ shr_n,
    const float* __restrict__ hz_n,  const float* __restrict__ wz_n,
    const float* __restrict__ hs_n,  const float* __restrict__ ws_n,
    float* __restrict__ out)
{
    const int lane  = threadIdx.x & 31;
    const int wave  = threadIdx.x >> 5;
    const int gwave = blockIdx.x * WAVES_PER_BLOCK + wave;   // one wave == one 16x16 tile
    const int b     = gwave >> 10;          // image index (gwave / 1024)
    const int tile  = gwave & 1023;
    const int r0    = (tile >> 5) << 4;     // tile row origin
    const int c0    = (tile & 31) << 4;     // tile col origin

    // ---- per-image affine matrix (wave-uniform; recomputed per lane, trivially cheap) ----
    const float PI = 3.14159265358979f;
    const float rot = PI * (15.0f * rot_n[b]) / 180.0f;
    const float shr = PI * ( 5.0f * shr_n[b]) / 180.0f;
    const float ihz = 1.0f / (1.0f + hz_n[b] * 0.1f);
    const float iwz = 1.0f / (1.0f + wz_n[b] * 0.1f);
    const float hs  = 16.0f * hs_n[b];
    const float ws  = 16.0f * ws_n[b];
    const float c1 = cosf(rot), s1 = sinf(rot);
    const float c2 = cosf(shr), s2 = sinf(shr);
    // A = rot_m @ shear_m ; Z = zoom_m @ shift_m ; m = A @ Z (only rows 0,1 needed)
    const float a01 =  c1 * s2 + s1 * c2;
    const float a11 = -s1 * s2 + c1 * c2;
    const float m00 =  c1 * ihz;
    const float m01 =  a01 * iwz;
    const float m02 =  c1 * hs * ihz + a01 * ws * iwz;
    const float m10 = -s1 * ihz;
    const float m11 =  a11 * iwz;
    const float m12 = -s1 * hs * ihz + a11 * ws * iwz;

    // ---- WMMA operand setup: D(16x16) = A(16x4) x B(4x16) in f32 ----
    // A[m][0] = x_m = 256 - (r0+m) ; A[m][1] = 1 ; A[m][2..3] = 0
    //   32-bit A layout: lanes 0-15 hold M=lane, VGPR0=K0, VGPR1=K1;
    //                    lanes 16-31 hold K2,K3 (zero here).
    // B[0][n] = m00 (broadcast) ; B[1][n] = m01*y_n + m02 ; y_n = (c0+n) - 256
    //   B layout: VGPR0: lanes 0-15 = row K0, lanes 16-31 = row K1; VGPR1 (K2,K3) = 0.
    const bool hi  = lane >= 16;
    const int  l16 = lane & 15;
    const float yc = (float)(c0 + l16 - HALF);

    v2f av;
    av.x = hi ? 0.0f : (float)(HALF - (r0 + l16));
    av.y = hi ? 0.0f : 1.0f;

    v2f bxv, byv;
    bxv.x = hi ? (m01 * yc + m02) : m00;  bxv.y = 0.0f;
    byv.x = hi ? (m11 * yc + m12) : m10;  byv.y = 0.0f;

    v8f czero = {};
    v8f xs, ys;
#if __has_builtin(__builtin_amdgcn_wmma_f32_16x16x4_f32)
    // 8 args: (neg_a, A, neg_b, B, c_mod, C, reuse_a, reuse_b)
    xs = __builtin_amdgcn_wmma_f32_16x16x4_f32(false, av, false, bxv, (short)0, czero, false, false);
    ys = __builtin_amdgcn_wmma_f32_16x16x4_f32(false, av, false, byv, (short)0, czero, false, false);
#else
    // VALU fallback producing the identical per-lane D fragment
    #pragma unroll
    for (int v = 0; v < 8; ++v) {
        const int   M  = v + (hi ? 8 : 0);
        const float xm = (float)(HALF - (r0 + M));
        xs[v] = m00 * xm + m01 * yc + m02;
        ys[v] = m10 * xm + m11 * yc + m12;
    }
#endif

    // ---- decode D fragment, clip, gather, store ----
    // D layout (32-bit 16x16): lane L, VGPR v -> M = v + 8*(L>=16), N = L&15
    const Pix3* __restrict__ img = (const Pix3*)images + (size_t)b * (DIM * DIM);
    Pix3*       __restrict__ dst = (Pix3*)out          + (size_t)b * (DIM * DIM);

    #pragma unroll
    for (int v = 0; v < 8; ++v) {
        const int M = v + (hi ? 8 : 0);
        int xi = (int)xs[v];                          // truncation toward zero (matches astype)
        int yi = (int)ys[v];
        xi = min(max(xi, -HALF + 1), HALF);           // clip to [-255, 256]
        yi = min(max(yi, -HALF + 1), HALF);
        const int srow = HALF - xi;                   // 0..511
        const int scol = HALF - 1 + yi;               // 0..511
        const Pix3 p = img[srow * DIM + scol];        // 12B gather (b96)
        dst[(r0 + M) * DIM + (c0 + l16)] = p;         // coalesced 12B store
    }
}

extern "C" void kernel_launch(void* const* d_in, const int* in_sizes, int n_in,
                              void* d_out, int out_size, void* d_ws, size_t ws_size,
                              hipStream_t stream) {
    const float* images = (const float*)d_in[0];
    const float* rot_n  = (const float*)d_in[1];
    const float* shr_n  = (const float*)d_in[2];
    const float* hz_n   = (const float*)d_in[3];
    const float* wz_n   = (const float*)d_in[4];
    const float* hs_n   = (const float*)d_in[5];
    const float* ws_n   = (const float*)d_in[6];
    float* out = (float*)d_out;

    const int B = in_sizes[1];                              // 64 images
    const int nblocks = (B * TILES_PER_IMG) / WAVES_PER_BLOCK;  // 8192 blocks of 256 threads
    Data_Augmentation_warp_kernel<<<nblocks, 256, 0, stream>>>(
        images, rot_n, shr_n, hz_n, wz_n, hs_n, ws_n, out);
}